// BarycenterClassification_2791728742479
// MI455X (gfx1250) — compile-verified
//
#include <hip/hip_runtime.h>
#include <hip/hip_bf16.h>

// ---------------------------------------------------------------------------
// BarycenterClassification loss for MI455X (gfx1250, wave32, WMMA).
// One 32x32 SPD matrix per wave: lane j owns column j. Matrices live in LDS.
// All 32x32x32 f32 matmuls go through V_WMMA_F32_16X16X4_F32.
// ---------------------------------------------------------------------------

#define B_SAMPLES 2048
#define NC 8
#define NSWEEP 8
#define NSWEEP_D 6
#define EV_EPS 1e-10f
#define CHOL_EPS 1e-3f
#define LAMBDA1 0.001f

typedef float v2f __attribute__((ext_vector_type(2)));
typedef float v8f __attribute__((ext_vector_type(8)));

// ---------------------------------------------------------------------------
// C(32x32) = A(32x32) * B(32x32)  (row-major, stride 32).  A/B may be generic
// (LDS or global) pointers; C must be writable (LDS).  transB reads B^T.
// Uses V_WMMA_F32_16X16X4_F32: D(16x16) = A(16x4) x B(4x16) + C.
// Fragment layouts per CDNA5 ISA 7.12.2 (wave32).
// ---------------------------------------------------------------------------
__device__ __forceinline__ void wave_mm32(const float* __restrict__ A,
                                          const float* __restrict__ Bm,
                                          float* __restrict__ C,
                                          int lane, bool transB) {
  const int m16 = lane & 15;
  const int hi  = lane >> 4;   // 0: lanes 0-15, 1: lanes 16-31
#pragma unroll
  for (int ti = 0; ti < 2; ++ti) {
#pragma unroll
    for (int tj = 0; tj < 2; ++tj) {
      v8f acc = {0.f, 0.f, 0.f, 0.f, 0.f, 0.f, 0.f, 0.f};
#pragma unroll
      for (int k4 = 0; k4 < 32; k4 += 4) {
        // A fragment (16x4): lanes 0-15 hold K=k4+0,k4+1; lanes 16-31 hold K=k4+2,k4+3
        const int am = ti * 16 + m16;
        const int ak = k4 + hi * 2;
        v2f a;
        a.x = A[am * 32 + ak];
        a.y = A[am * 32 + ak + 1];
        // B fragment (4x16): VGPR0 rows k4+hi ; VGPR1 rows k4+hi+2, cols tj*16+m16
        const int bn  = tj * 16 + m16;
        const int bk0 = k4 + hi;
        v2f b;
        if (!transB) {
          b.x = Bm[bk0 * 32 + bn];
          b.y = Bm[(bk0 + 2) * 32 + bn];
        } else {
          b.x = Bm[bn * 32 + bk0];
          b.y = Bm[bn * 32 + bk0 + 2];
        }
        acc = __builtin_amdgcn_wmma_f32_16x16x4_f32(
            /*neg_a=*/false, a, /*neg_b=*/false, b,
            /*c_mod=*/(short)0, acc, /*reuse_a=*/false, /*reuse_b=*/false);
      }
      // D layout: VGPR r -> row ti*16 + r + 8*hi, col tj*16 + m16
#pragma unroll
      for (int r = 0; r < 8; ++r) {
        C[(ti * 16 + r + 8 * hi) * 32 + tj * 16 + m16] = acc[r];
      }
    }
  }
}

// ---------------------------------------------------------------------------
// Cyclic two-sided Jacobi eigensolver on a 32x32 symmetric matrix in LDS.
// lane j updates column/row j.  Relies on in-order whole-wave DS semantics.
// On exit: diag(A) = eigenvalues, V = eigenvectors (columns) if wantV.
// ---------------------------------------------------------------------------
__device__ void wave_jacobi(float* A, float* V, int lane, bool wantV, int sweeps) {
  if (wantV) {
#pragma unroll
    for (int i = 0; i < 32; ++i) V[i * 32 + lane] = (i == lane) ? 1.f : 0.f;
  }
#pragma unroll 1
  for (int sw = 0; sw < sweeps; ++sw) {
#pragma unroll 1
    for (int p = 0; p < 31; ++p) {
#pragma unroll 1
      for (int q = p + 1; q < 32; ++q) {
        const float apq = A[p * 32 + q];   // wave-uniform scalar
        const float app = A[p * 32 + p];
        const float aqq = A[q * 32 + q];
        const float aapq = fabsf(apq);
        if (aapq <= 1e-12f * (fabsf(app) + fabsf(aqq)) + 1e-32f) continue;
        const float tau = (aqq - app) / (2.0f * apq);
        const float t   = copysignf(1.0f, tau) / (fabsf(tau) + sqrtf(1.0f + tau * tau));
        const float c   = 1.0f / sqrtf(1.0f + t * t);
        const float s   = t * c;
        // rows p,q  (A <- G^T A)
        float rp = A[p * 32 + lane], rq = A[q * 32 + lane];
        A[p * 32 + lane] = c * rp - s * rq;
        A[q * 32 + lane] = s * rp + c * rq;
        // cols p,q  (A <- A G)
        float cp = A[lane * 32 + p], cq = A[lane * 32 + q];
        A[lane * 32 + p] = c * cp - s * cq;
        A[lane * 32 + q] = s * cp + c * cq;
        if (wantV) {
          float vp = V[lane * 32 + p], vq = V[lane * 32 + q];
          V[lane * 32 + p] = c * vp - s * vq;
          V[lane * 32 + q] = s * vp + c * vq;
        }
      }
    }
  }
}

__device__ __forceinline__ float lane_sum32(float v) {
#pragma unroll
  for (int m = 16; m > 0; m >>= 1) v += __shfl_xor(v, m, 32);
  return v;
}

// ---------------------------------------------------------------------------
// K0: zero scratch regions used as accumulators
// ---------------------------------------------------------------------------
__global__ void k_zero(float* p, int n) {
  int i = blockIdx.x * blockDim.x + threadIdx.x;
  if (i < n) p[i] = 0.f;
}

// ---------------------------------------------------------------------------
// K1: per-class arithmetic means + member counts.  blockIdx = class,
// thread t = matrix element (1024 threads).
// ---------------------------------------------------------------------------
__global__ void k_classmean(const float* __restrict__ X,
                            const int* __restrict__ labels,
                            float* __restrict__ bary0,
                            float* __restrict__ counts) {
  const int c = blockIdx.x;
  const int t = threadIdx.x;
  float s = 0.f;
  int cnt = 0;
  for (int b = 0; b < B_SAMPLES; ++b) {
    if (labels[b] == c) { s += X[b * 1024 + t]; ++cnt; }
  }
  const float inv = 1.0f / (float)(cnt > 0 ? cnt : 1);
  bary0[c * 1024 + t] = s * inv;
  if (t == 0) counts[c] = (float)(cnt > 0 ? cnt : 1);
}

// ---------------------------------------------------------------------------
// K2: per-class eigh(bary0) -> sqrtm, invsqrtm.  One class per wave.
// ---------------------------------------------------------------------------
__global__ void k_sqrtm(const float* __restrict__ bary0,
                        float* __restrict__ sqm,
                        float* __restrict__ isq) {
  __shared__ float sm[2][4][1024];   // 32 KB
  const int wave = threadIdx.x >> 5;
  const int lane = threadIdx.x & 31;
  const int c = blockIdx.x * 2 + wave;
  float* A  = sm[wave][0];
  float* V  = sm[wave][1];
  float* T1 = sm[wave][2];
  float* T2 = sm[wave][3];
#pragma unroll
  for (int i = 0; i < 32; ++i) A[i * 32 + lane] = bary0[c * 1024 + i * 32 + lane];
  wave_jacobi(A, V, lane, true, NSWEEP);
  const float s  = fmaxf(A[lane * 32 + lane], 1e-20f);
  const float fs = sqrtf(s);
  const float fi = 1.0f / fs;
  // sqrtm = (U * diag(sqrt s)) @ U^T
#pragma unroll
  for (int i = 0; i < 32; ++i) T1[i * 32 + lane] = V[i * 32 + lane] * fs;
  wave_mm32(T1, V, T2, lane, /*transB=*/true);
#pragma unroll
  for (int i = 0; i < 32; ++i) sqm[c * 1024 + i * 32 + lane] = T2[i * 32 + lane];
  // invsqrtm = (U * diag(1/sqrt s)) @ U^T
#pragma unroll
  for (int i = 0; i < 32; ++i) T1[i * 32 + lane] = V[i * 32 + lane] * fi;
  wave_mm32(T1, V, T2, lane, /*transB=*/true);
#pragma unroll
  for (int i = 0; i < 32; ++i) isq[c * 1024 + i * 32 + lane] = T2[i * 32 + lane];
}

// ---------------------------------------------------------------------------
// K3: per-sample tangent vector accumulated into meanT[labels[b]].
// Only class members carry nonzero weight in the reference einsum.
// ---------------------------------------------------------------------------
__global__ void k_tangent(const float* __restrict__ X,
                          const int* __restrict__ labels,
                          const float* __restrict__ sqm,
                          const float* __restrict__ isq,
                          const float* __restrict__ counts,
                          float* __restrict__ meanT) {
  __shared__ float sm[2][4][1024];
  const int wave = threadIdx.x >> 5;
  const int lane = threadIdx.x & 31;
  const int b = blockIdx.x * 2 + wave;
  const int c = labels[b];                // wave-uniform
  const float* Xb = X + b * 1024;
  const float* sq = sqm + c * 1024;
  const float* iq = isq + c * 1024;
  float* M  = sm[wave][0];
  float* V  = sm[wave][1];
  float* T1 = sm[wave][2];
  float* T2 = sm[wave][3];
  // M = invsqrtm @ X @ invsqrtm
  wave_mm32(iq, Xb, T1, lane, false);
  wave_mm32(T1, iq, M, lane, false);
  wave_jacobi(M, V, lane, true, NSWEEP);
  // logM = V diag(log(max(ev,eps))) V^T
  const float l = logf(fmaxf(M[lane * 32 + lane], EV_EPS));
#pragma unroll
  for (int i = 0; i < 32; ++i) T1[i * 32 + lane] = V[i * 32 + lane] * l;
  wave_mm32(T1, V, T2, lane, /*transB=*/true);
  // T = sqrtm @ logM @ sqrtm
  wave_mm32(sq, T2, T1, lane, false);
  wave_mm32(T1, sq, T2, lane, false);
  const float wgt = 1.0f / counts[c];
#pragma unroll
  for (int i = 0; i < 32; ++i)
    atomicAdd(&meanT[c * 1024 + i * 32 + lane], T2[i * 32 + lane] * wgt);
}

// ---------------------------------------------------------------------------
// K4: per-class exponential update + convergence quirk.
// ---------------------------------------------------------------------------
__global__ void k_update(const float* __restrict__ bary0,
                         const float* __restrict__ sqm,
                         const float* __restrict__ isq,
                         const float* __restrict__ meanT,
                         float* __restrict__ bary) {
  __shared__ float sm[2][4][1024];
  const int wave = threadIdx.x >> 5;
  const int lane = threadIdx.x & 31;
  const int c = blockIdx.x * 2 + wave;
  const float* iq = isq + c * 1024;
  const float* sq = sqm + c * 1024;
  const float* mt = meanT + c * 1024;
  float* G0 = sm[wave][0];
  float* G1 = sm[wave][1];
  float* G2 = sm[wave][2];
  float* G3 = sm[wave][3];
  // N = invsqrtm @ meanT @ invsqrtm
  wave_mm32(iq, mt, G0, lane, false);
  wave_mm32(G0, iq, G1, lane, false);
  wave_jacobi(G1, G3, lane, true, NSWEEP);
  // expN = Vn diag(exp(max(en, eps))) Vn^T   (faithful clamp-before-exp quirk)
  const float e = expf(fmaxf(G1[lane * 32 + lane], EV_EPS));
#pragma unroll
  for (int i = 0; i < 32; ++i) G2[i * 32 + lane] = G3[i * 32 + lane] * e;
  wave_mm32(G2, G3, G0, lane, /*transB=*/true);
  // new = sqrtm @ expN @ sqrtm
  wave_mm32(sq, G0, G1, lane, false);
  wave_mm32(G1, sq, G2, lane, false);
  // err = ||new - bary0||_F ; if err < tol return bary0 unchanged
  float err2 = 0.f;
#pragma unroll
  for (int i = 0; i < 32; ++i) {
    const float d = G2[i * 32 + lane] - bary0[c * 1024 + i * 32 + lane];
    err2 += d * d;
  }
  err2 = lane_sum32(err2);
  const bool conv = err2 < (1e-4f * 1e-4f);
#pragma unroll
  for (int i = 0; i < 32; ++i)
    bary[c * 1024 + i * 32 + lane] =
        conv ? bary0[c * 1024 + i * 32 + lane] : G2[i * 32 + lane];
}

// ---------------------------------------------------------------------------
// K5: per-sample AIRM distances to all 8 barycenters.
// Cholesky + triangular inverse per wave, then M = L^{-T} bary L^{-T}
// (faithful asymmetric quirk), symmetrize from lower triangle, eigenvalues.
// ---------------------------------------------------------------------------
__global__ void k_dist(const float* __restrict__ X,
                       const float* __restrict__ bary,
                       float* __restrict__ D) {
  __shared__ float sm[2][4][1024];
  const int wave = threadIdx.x >> 5;
  const int lane = threadIdx.x & 31;
  const int b = blockIdx.x * 2 + wave;
  float* B0 = sm[wave][0];   // X+epsI -> L (lower)
  float* B1 = sm[wave][1];   // L^{-1}, then scratch
  float* B2 = sm[wave][2];   // LT = (L^{-1})^T, preserved over class loop
  float* B3 = sm[wave][3];   // symmetrized M for Jacobi
#pragma unroll
  for (int i = 0; i < 32; ++i)
    B0[i * 32 + lane] = X[b * 1024 + i * 32 + lane] + ((i == lane) ? CHOL_EPS : 0.f);
  // In-place Cholesky (lower triangle); lane i owns row i.
#pragma unroll 1
  for (int j = 0; j < 32; ++j) {
    const float d = sqrtf(fmaxf(B0[j * 32 + j], 1e-30f));
    const float aij = B0[lane * 32 + j];
    if (lane == j)      B0[lane * 32 + j] = d;
    else if (lane > j)  B0[lane * 32 + j] = aij / d;
    if (lane > j) {
      const float lij = B0[lane * 32 + j];
      for (int k = j + 1; k <= lane; ++k)
        B0[lane * 32 + k] -= lij * B0[k * 32 + j];
    }
  }
  // B1 = L^{-1}: lane j solves L y = e_j by forward substitution.
#pragma unroll 1
  for (int i = 0; i < 32; ++i) {
    float s = (i == lane) ? 1.f : 0.f;
    for (int k = lane; k < i; ++k) s -= B0[i * 32 + k] * B1[k * 32 + lane];
    B1[i * 32 + lane] = (i >= lane) ? s / B0[i * 32 + i] : 0.f;
  }
  // B2 = (L^{-1})^T
#pragma unroll
  for (int i = 0; i < 32; ++i) B2[i * 32 + lane] = B1[lane * 32 + i];
#pragma unroll 1
  for (int c = 0; c < NC; ++c) {
    // M = LT @ bary_c @ LT
    wave_mm32(B2, bary + c * 1024, B0, lane, false);
    wave_mm32(B0, B2, B1, lane, false);
    // eigvalsh(UPLO='L'): symmetrize from the lower triangle
#pragma unroll
    for (int i = 0; i < 32; ++i)
      B3[i * 32 + lane] = (i >= lane) ? B1[i * 32 + lane] : B1[lane * 32 + i];
    wave_jacobi(B3, B3, lane, /*wantV=*/false, NSWEEP_D);
    const float l = logf(fmaxf(B3[lane * 32 + lane], EV_EPS));
    const float s2 = lane_sum32(l * l);
    if (lane == 0) D[b * NC + c] = sqrtf(s2);
  }
}

// ---------------------------------------------------------------------------
// K6: final scalar loss.
// ---------------------------------------------------------------------------
__global__ void k_loss(const float* __restrict__ D,
                       const float* __restrict__ out,
                       const int* __restrict__ labels,
                       float* __restrict__ res) {
  __shared__ float s1[256], s2[256], s3[256];
  const int t = threadIdx.x;
  float intraSum = 0.f, dispSum = 0.f, ceSum = 0.f;
  for (int b = t; b < B_SAMPLES; b += 256) {
    const int lab = labels[b];
    float m = -3.0e38f;
#pragma unroll
    for (int c = 0; c < NC; ++c) m = fmaxf(m, out[b * NC + c]);
    float se = 0.f;
#pragma unroll
    for (int c = 0; c < NC; ++c) se += expf(out[b * NC + c] - m);
    ceSum += m + logf(se) - out[b * NC + lab];
#pragma unroll
    for (int c = 0; c < NC; ++c) {
      const float d = D[b * NC + c];
      if (c == lab) intraSum += d; else dispSum += d;
    }
  }
  s1[t] = intraSum; s2[t] = dispSum; s3[t] = ceSum;
  __syncthreads();
  for (int o = 128; o > 0; o >>= 1) {
    if (t < o) { s1[t] += s1[t + o]; s2[t] += s2[t + o]; s3[t] += s3[t + o]; }
    __syncthreads();
  }
  if (t == 0) {
    const float intra = LAMBDA1 * s1[0] / (float)B_SAMPLES;
    const float disp  = s2[0] / ((float)B_SAMPLES * (float)(NC - 1));
    const float ce    = s3[0] / (float)B_SAMPLES;
    res[0] = intra - LAMBDA1 * disp + ce;
  }
}

// ---------------------------------------------------------------------------
extern "C" void kernel_launch(void* const* d_in, const int* in_sizes, int n_in,
                              void* d_out, int out_size, void* d_ws, size_t ws_size,
                              hipStream_t stream) {
  const float* X      = (const float*)d_in[0];   // (2048,32,32)
  const float* logits = (const float*)d_in[1];   // (2048,8)
  const int*   labels = (const int*)d_in[2];     // (2048,)

  float* w      = (float*)d_ws;
  float* counts = w;                 // 16 floats (8 used)
  float* bary0  = w + 16;            // 8*1024
  float* sqm    = bary0 + NC * 1024; // 8*1024
  float* isq    = sqm   + NC * 1024; // 8*1024
  float* meanT  = isq   + NC * 1024; // 8*1024 (atomic accum -> must zero)
  float* bary   = meanT + NC * 1024; // 8*1024
  float* D      = bary  + NC * 1024; // 2048*8

  k_zero<<<(NC * 1024 + 255) / 256, 256, 0, stream>>>(meanT, NC * 1024);
  k_classmean<<<NC, 1024, 0, stream>>>(X, labels, bary0, counts);
  k_sqrtm<<<NC / 2, 64, 0, stream>>>(bary0, sqm, isq);
  k_tangent<<<B_SAMPLES / 2, 64, 0, stream>>>(X, labels, sqm, isq, counts, meanT);
  k_update<<<NC / 2, 64, 0, stream>>>(bary0, sqm, isq, meanT, bary);
  k_dist<<<B_SAMPLES / 2, 64, 0, stream>>>(X, bary, D);
  k_loss<<<1, 256, 0, stream>>>(D, logits, labels, (float*)d_out);
}